// MARLLoss_72885595013508
// MI455X (gfx1250) — compile-verified
//
#include <hip/hip_runtime.h>
#include <math.h>

typedef __attribute__((ext_vector_type(2))) float v2f;
typedef __attribute__((ext_vector_type(8))) float v8f;

#define IMG_B 8
#define IMG_H 512
#define IMG_W 512
#define WAVES_PER_BLOCK 8
#define NTILES (IMG_B * (IMG_H/16) * (IMG_W/16))   // 8192 tiles, one wave each
#define NBLOCKS (NTILES / WAVES_PER_BLOCK)         // 1024
#define VCOLS 50                                   // padded V scratch width (8B-aligned rows, bank-skewed)

// Branch-free bounds-checked image load: clamp address, select 0 if OOB.
__device__ __forceinline__ float ldimg(const float* __restrict__ img, int y, int x) {
  const int yc = min(max(y, 0), IMG_H - 1);
  const int xc = min(max(x, 0), IMG_W - 1);
  const float v = img[(size_t)yc * IMG_W + xc];
  const bool inb = ((unsigned)y < (unsigned)IMG_H) & ((unsigned)x < (unsigned)IMG_W);
  return inb ? v : 0.0f;
}

// Separable SAME conv of one 16x16 tile on the f32 WMMA pipe.
//   vertical:   V(16 x P)  = Toeplitz(w)(16 x P) * In(P x P)    [A = const band, B from global]
//   horizontal: Out(16x16) = V(16 x P) * Toeplitz(w)(P x 16)    [A via LDS relayout, B = const band]
// The vertical-A and horizontal-B band fragments are identical per lane -> computed once.
// INTERIOR: tile + halo fully in-image -> unconditional loads with immediate offsets.
template<int R, bool INTERIOR>
__device__ __forceinline__ v8f conv_sep_wmma(const float* __restrict__ img,
                                             int y0, int x0,
                                             const float* w,            // LDS taps, 2R+1
                                             float (*vbuf)[VCOLS],      // per-wave LDS scratch
                                             int l15, int half) {
  constexpr int P   = 16 + 2 * R;
  constexpr int KP  = (P + 3) & ~3;        // K padded to multiple of 4
  constexpr int NCH = (P + 15) / 16;       // N chunks for the vertical pass
  constexpr int NK  = KP / 4;              // WMMA K-steps

  // Banded weight fragments (branch-free: clamped LDS index + select-zero).
  v2f wf[NK];
  #pragma unroll
  for (int k = 0; k < NK; ++k) {
    const int d0 = 4 * k + 2 * half - l15;
    const int c0 = min(max(d0, 0), 2 * R);
    const int c1 = min(max(d0 + 1, 0), 2 * R);
    const float w0 = w[c0];
    const float w1 = w[c1];
    wf[k].x = ((unsigned)d0       <= (unsigned)(2 * R)) ? w0 : 0.0f;
    wf[k].y = ((unsigned)(d0 + 1) <= (unsigned)(2 * R)) ? w1 : 0.0f;
  }

  // ---------------- vertical pass ----------------
  #pragma unroll
  for (int nc = 0; nc < NCH; ++nc) {
    v8f acc = {};
    const int xg = x0 - R + nc * 16 + l15;
    const int yb = y0 - R + 2 * half;
    if constexpr (INTERIOR) {
      // One base address per chunk; all K-steps use compile-time immediate offsets.
      const float* __restrict__ p = img + (size_t)yb * IMG_W + xg;
      #pragma unroll
      for (int k = 0; k < NK; ++k) {
        v2f bm;
        bm.x = p[(4 * k)     * IMG_W];
        bm.y = p[(4 * k + 1) * IMG_W];
        acc = __builtin_amdgcn_wmma_f32_16x16x4_f32(false, wf[k], false, bm,
                                                    (short)0, acc, false, false);
      }
    } else {
      #pragma unroll
      for (int k = 0; k < NK; ++k) {
        v2f bm;
        bm.x = ldimg(img, yb + 4 * k,     xg);
        bm.y = ldimg(img, yb + 4 * k + 1, xg);
        acc = __builtin_amdgcn_wmma_f32_16x16x4_f32(false, wf[k], false, bm,
                                                    (short)0, acc, false, false);
      }
    }
    // spill V chunk (C/D layout -> [row][col] in LDS)
    #pragma unroll
    for (int vj = 0; vj < 8; ++vj)
      vbuf[vj + 8 * half][nc * 16 + l15] = acc[vj];
  }

  // ---------------- horizontal pass ----------------
  v8f out = {};
  const float* vrow = &vbuf[l15][2 * half];     // 8B-aligned (VCOLS even, 2*half even)
  #pragma unroll
  for (int k = 0; k < NK; ++k) {
    const v2f a = *(const v2f*)(vrow + 4 * k);  // ds_load_b64: K = 4k+2h, 4k+2h+1
    out = __builtin_amdgcn_wmma_f32_16x16x4_f32(false, a, false, wf[k],
                                                (short)0, out, false, false);
  }
  return out;
}

__global__ void marl_zero_kernel(float* __restrict__ out) {
  if (threadIdx.x == 0 && blockIdx.x == 0) out[0] = 0.0f;
}

__global__ void marl_reduce_kernel(const float* __restrict__ part, float* __restrict__ out) {
  __shared__ float sm[256];
  float s = 0.0f;
  for (int i = threadIdx.x; i < NBLOCKS; i += 256) s += part[i];  // fixed order
  sm[threadIdx.x] = s;
  __syncthreads();
  #pragma unroll
  for (int off = 128; off > 0; off >>= 1) {
    if ((int)threadIdx.x < off) sm[threadIdx.x] += sm[threadIdx.x + off];
    __syncthreads();
  }
  if (threadIdx.x == 0) out[0] = sm[0] * (-1.0f / (float)IMG_B);
}

__global__ __launch_bounds__(WAVES_PER_BLOCK * 32)
void MARLLoss_72885595013508_kernel(const float* __restrict__ prob,
                                    const float* __restrict__ cimg,
                                    const float* __restrict__ himg,
                                    float* __restrict__ partial,
                                    float* __restrict__ out,
                                    int use_atomic) {
  __shared__ float w11[16];
  __shared__ float w21[24];
  __shared__ float sumk2_s;
  __shared__ __align__(16) float Vbuf[WAVES_PER_BLOCK][16][VCOLS];
  __shared__ float wsum[WAVES_PER_BLOCK];

  // Normalized 1D Gaussian (11 taps), its self-convolution (21 taps = 1D of G*G),
  // and sum(k^2) = (sum g_i^2)^2 for the separable 2D kernel.
  if (threadIdx.x == 0) {
    float g[11];
    float s = 0.0f;
    for (int i = 0; i < 11; ++i) {
      const float d = (float)i - 5.0f;
      g[i] = __expf(-(d * d) / 8.0f);      // 2*sigma^2 = 8
      s += g[i];
    }
    const float inv = 1.0f / s;
    float s2 = 0.0f;
    for (int i = 0; i < 11; ++i) {
      g[i] *= inv;
      w11[i] = g[i];
      s2 += g[i] * g[i];
    }
    sumk2_s = s2 * s2;
    for (int t = 0; t < 21; ++t) {
      float acc = 0.0f;
      for (int i = 0; i < 11; ++i) {
        const int j = t - i;
        if (j >= 0 && j <= 10) acc += g[i] * g[j];
      }
      w21[t] = acc;
    }
  }
  __syncthreads();

  const int warp = threadIdx.x >> 5;
  const int lane = threadIdx.x & 31;
  const int half = lane >> 4;
  const int l15  = lane & 15;

  const int wid = blockIdx.x * WAVES_PER_BLOCK + warp;  // one wave per 16x16 tile
  const int t   = wid & 1023;                           // 1024 tiles per image
  const int y0  = (t >> 5) << 4;
  const int x0  = (t & 31) << 4;

  const size_t base = (size_t)(wid >> 10) * IMG_H * IMG_W;
  const float* __restrict__ hb = himg + base;
  const float* __restrict__ cb = cimg + base;
  const float* __restrict__ pb = prob + base;

  float (*vbuf)[VCOLS] = Vbuf[warp];

  // corr = (G*G)*h - G*c. Interior tiles (88%) take the checkless fast path.
  const int interior = (y0 >= 16) & (y0 <= IMG_H - 32) & (x0 >= 16) & (x0 <= IMG_W - 32);
  v8f corrH, corrC;
  if (__builtin_amdgcn_readfirstlane(interior)) {     // wave-uniform scalar branch
    corrH = conv_sep_wmma<10, true >(hb, y0, x0, w21, vbuf, l15, half);
    corrC = conv_sep_wmma<5,  true >(cb, y0, x0, w11, vbuf, l15, half);
  } else {
    corrH = conv_sep_wmma<10, false>(hb, y0, x0, w21, vbuf, l15, half);
    corrC = conv_sep_wmma<5,  false>(cb, y0, x0, w11, vbuf, l15, half);
  }

  // Epilogue: reward * logp, summed. Base pointer + immediate row offsets.
  const size_t eoff = (size_t)(y0 + 8 * half) * IMG_W + (x0 + l15);
  const float* __restrict__ hp = hb + eoff;
  const float* __restrict__ pp = pb + eoff;
  const float sk2 = sumk2_s;
  float local = 0.0f;
  #pragma unroll
  for (int vj = 0; vj < 8; ++vj) {
    const float hv = hp[vj * IMG_W];
    const float pv = pp[vj * IMG_W];
    const float corr   = corrH[vj] - corrC[vj];
    const float delta  = 1.0f - 2.0f * hv;
    const float reward = 2.0f * delta * corr + delta * delta * sk2;
    const float lp = (hv > 0.5f) ? __logf(pv + 1e-8f) : __logf(1.0f - pv + 1e-8f);
    local += reward * lp;
  }

  // Deterministic reduction: fixed-order wave shuffle, then fixed-order cross-wave sum.
  #pragma unroll
  for (int off = 16; off > 0; off >>= 1)
    local += __shfl_down(local, off, 32);
  if (lane == 0) wsum[warp] = local;
  __syncthreads();
  if (threadIdx.x == 0) {
    float s = 0.0f;
    #pragma unroll
    for (int i = 0; i < WAVES_PER_BLOCK; ++i) s += wsum[i];
    if (use_atomic) atomicAdd(out, s * (-1.0f / (float)IMG_B));
    else            partial[blockIdx.x] = s;
  }
}

extern "C" void kernel_launch(void* const* d_in, const int* in_sizes, int n_in,
                              void* d_out, int out_size, void* d_ws, size_t ws_size,
                              hipStream_t stream) {
  (void)in_sizes; (void)n_in; (void)out_size;
  const float* prob = (const float*)d_in[0];   // prob_map
  const float* cimg = (const float*)d_in[1];   // c
  const float* himg = (const float*)d_in[2];   // h_sampled
  float* out = (float*)d_out;                  // scalar loss

  if (ws_size >= NBLOCKS * sizeof(float)) {
    float* part = (float*)d_ws;
    MARLLoss_72885595013508_kernel<<<NBLOCKS, WAVES_PER_BLOCK * 32, 0, stream>>>(
        prob, cimg, himg, part, out, 0);
    marl_reduce_kernel<<<1, 256, 0, stream>>>(part, out);
  } else {
    // Fallback (tiny/absent workspace): atomic accumulation.
    marl_zero_kernel<<<1, 32, 0, stream>>>(out);
    MARLLoss_72885595013508_kernel<<<NBLOCKS, WAVES_PER_BLOCK * 32, 0, stream>>>(
        prob, cimg, himg, nullptr, out, 1);
  }
}